// GraphWNblocks_act_78383153152618
// MI455X (gfx1250) — compile-verified
//
#include <hip/hip_runtime.h>
#include <math.h>

#define BB   8
#define NN   300
#define TTF  48
#define TCC  47
#define CR   32
#define CS   256

#define NPOS   (BB*TCC*NN)        /* 112800 positions (b,t,n) */
#define XO_SZ  (BB*CR*NN*TCC)     /* 3,609,600 */
#define NROWT  19                 /* ceil(300/16) row tiles */
#define SASTR  40                 /* LDS row stride (floats): 2*40 % 64 == 16 -> conflict-free */
#define WSSTR  264                /* LDS stride for Ws rows (256 wide): 264 % 32 == 8 */
#define WLSTR  40                 /* LDS stride for Wl rows (32 wide) */

typedef float v2f __attribute__((ext_vector_type(2)));
typedef float v8f __attribute__((ext_vector_type(8)));
typedef unsigned int u32x4 __attribute__((ext_vector_type(4)));
typedef int i32x4 __attribute__((ext_vector_type(4)));
typedef int i32x8 __attribute__((ext_vector_type(8)));

#if defined(__has_builtin)
# if __has_builtin(__builtin_amdgcn_tensor_load_to_lds)
#  define HAVE_TDM 1
# endif
#endif
#if defined(__has_include)
# if __has_include(<hip/amd_detail/amd_gfx1250_TDM.h>)
#  define TDM_6ARG 1
# endif
#endif

__device__ __forceinline__ v8f wmma_f32(v2f a, v2f b, v8f c) {
  // V_WMMA_F32_16X16X4_F32 : D = A(16x4) * B(4x16) + C
  return __builtin_amdgcn_wmma_f32_16x16x4_f32(false, a, false, b, (short)0, c, false, false);
}

__device__ __forceinline__ void decode_step(int& b, int& t, int& n) {
  ++n;
  if (n == NN) { n = 0; ++t; if (t == TCC) { t = 0; ++b; } }
}

// ---------------------------------------------------------------------------
// Kernel 1: fused gated TCN -> v0 in (b,t,n,c) layout + phase index (b,t,n)
// ---------------------------------------------------------------------------
__global__ void k_tcn(const float* __restrict__ x, const float* __restrict__ xph,
                      const float* __restrict__ meanv, const float* __restrict__ stdv,
                      const float* __restrict__ Wf, const float* __restrict__ bf,
                      const float* __restrict__ Wg, const float* __restrict__ bg,
                      float* __restrict__ v0, int* __restrict__ ph)
{
  __shared__ float sWf[2048];
  __shared__ float sWg[2048];
  __shared__ float sx[8][64];
  for (int i = threadIdx.x; i < 2048; i += 256) { sWf[i] = Wf[i]; sWg[i] = Wg[i]; }
  const int wid = threadIdx.x >> 5, lane = threadIdx.x & 31;
  const int p = blockIdx.x * 8 + wid;                 // flat (b,t,n), exact
  const int b = p / (TCC*NN);
  const int r = p % (TCC*NN);
  const int t = r / NN;
  const int n = r % NN;
  const float* xb = x + (((size_t)(b*CR + lane)*NN + n)*TTF + t);
  sx[wid][lane]      = xb[0];
  sx[wid][lane + 32] = xb[1];
  __syncthreads();
  float f = bf[lane], g = bg[lane];
#pragma unroll
  for (int ci = 0; ci < 32; ++ci) {
    const float x0 = sx[wid][ci], x1 = sx[wid][ci + 32];
    f += sWf[(lane*32 + ci)*2] * x0 + sWf[(lane*32 + ci)*2 + 1] * x1;
    g += sWg[(lane*32 + ci)*2] * x0 + sWg[(lane*32 + ci)*2 + 1] * x1;
  }
  const float val = tanhf(f) * (1.0f / (1.0f + __expf(-g)));
  v0[(size_t)p * 32 + lane] = val;
  if (lane == 0) {
    int pv = 0;
    for (int k = 7; k >= 0; --k) {                    // first-true wins (argmax semantics)
      const float rr = xph[((size_t)(b*NN + n)*11 + 3 + k)*TCC + t] * stdv[3 + k] + meanv[3 + k];
      if (fabsf(rr - 1.0f) < 1e-6f) pv = k;
    }
    ph[p] = pv;
  }
}

// ---------------------------------------------------------------------------
// Kernel 1b: precompute phase-masked supports  Smask[sup][p][i][j]
// ---------------------------------------------------------------------------
__global__ void k_mask(const float* __restrict__ s1, const float* __restrict__ s2,
                       const int* __restrict__ adjph, float* __restrict__ Smask)
{
  const int idx = blockIdx.x * 256 + threadIdx.x;
  if (idx >= 2*NN*NN) return;
  const int sup = idx / (NN*NN), ij = idx % (NN*NN);
  const float v = (sup ? s2 : s1)[ij];
  const int ap = adjph[ij];
  float* o = Smask + (size_t)sup*8*NN*NN + ij;
#pragma unroll
  for (int p = 0; p < 8; ++p) o[(size_t)p*NN*NN] = (ap == p) ? v : 0.0f;
}

// ---------------------------------------------------------------------------
// Kernel 2: skip_out = v0 @ Ws^T + bs + skip   (WMMA f32, M=112800 K=32 N=256)
// Ws staged transposed in LDS with bank-conflict-free stride.
// ---------------------------------------------------------------------------
__global__ void k_skip(const float* __restrict__ v0, const float* __restrict__ Ws,
                       const float* __restrict__ bs, const float* __restrict__ skip,
                       float* __restrict__ oskip)
{
  __shared__ float sWs[32 * WSSTR];                   // [k][o], stride 264
  for (int i = threadIdx.x; i < CS*32; i += 256) {
    const int o = i >> 5, k = i & 31;
    sWs[k*WSSTR + o] = Ws[i];
  }
  __syncthreads();
  const int wid  = threadIdx.x >> 5, lane = threadIdx.x & 31;
  const int o0   = wid * 32;                          // this wave's 32 output channels
  const int row0 = blockIdx.x * 16;                   // 7050 row tiles of 16
  const int m = lane & 15, khalf = lane >> 4, col = lane & 15;
  const float* arow = v0 + (size_t)(row0 + m) * 32;
  v8f acc0 = {}, acc1 = {};
#pragma unroll
  for (int k0 = 0; k0 < 32; k0 += 4) {
    const int kk = k0 + 2 * khalf;
    v2f a;  a.x = arow[kk];  a.y = arow[kk + 1];
    v2f b0, b1;
    b0.x = sWs[kk*WSSTR + o0 + col];            b0.y = sWs[(kk+1)*WSSTR + o0 + col];
    b1.x = sWs[kk*WSSTR + o0 + 16 + col];       b1.y = sWs[(kk+1)*WSSTR + o0 + 16 + col];
    acc0 = wmma_f32(a, b0, acc0);
    acc1 = wmma_f32(a, b1, acc1);
  }
  const int o = o0 + col;
  const float bsv0 = bs[o], bsv1 = bs[o + 16];
  // incremental (b,t,n) decode for the 8 consecutive store rows of this lane
  int rb = row0 + 8*khalf;
  int b2 = rb / (TCC*NN); int rr = rb % (TCC*NN); int t2 = rr / NN; int n2 = rr % NN;
#pragma unroll
  for (int r = 0; r < 8; ++r) {
    const size_t i0 = (((size_t)(b2*CS + o     ))*NN + n2)*TCC + t2;
    const size_t i1 = (((size_t)(b2*CS + o + 16))*NN + n2)*TCC + t2;
    oskip[i0] = acc0[r] + bsv0 + skip[i0];
    oskip[i1] = acc1[r] + bsv1 + skip[i1];
    decode_step(b2, t2, n2);
  }
}

// ---------------------------------------------------------------------------
// Kernel 3: fused double diffusion hop for one (b,t,support):
//   v1 = Sm @ v0 + v0 ; v2 = Sm @ v1 + v1      (Sm row i = Smask[ph[i]][i][:])
// One 19-wave block per (b,t); v0/v1 staged in LDS (stride-40 padded rows),
// v0 stage via Tensor Data Mover when available.
// ---------------------------------------------------------------------------
__global__ void __launch_bounds__(608)
k_hop2(const float* __restrict__ Smask, const int* __restrict__ ph,
       const float* __restrict__ v0,
       float* __restrict__ w1, float* __restrict__ w2,
       float* __restrict__ w3, float* __restrict__ w4)
{
  __shared__ float sA[NN * SASTR];                    // v0 tile, padded rows
  __shared__ float sB[NN * SASTR];                    // v1 tile, padded rows
  const int bt  = blockIdx.x;
  const int sup = blockIdx.y;
  float* __restrict__ wo1 = sup ? w3 : w1;
  float* __restrict__ wo2 = sup ? w4 : w2;
  const float* __restrict__ Sb = Smask + (size_t)sup * 8 * NN * NN;
  const float* __restrict__ vsrc = v0 + (size_t)bt * NN * 32;

#ifdef HAVE_TDM
  if (threadIdx.x < 32) {
    const unsigned lds_off = (unsigned)(size_t)(void*)sA;
    const unsigned long long ga = (unsigned long long)(const void*)vsrc;
    u32x4 g0;
    g0[0] = 1u;                                       // count=1, user descriptor
    g0[1] = lds_off;                                  // D#.lds_addr
    g0[2] = (unsigned)ga;                             // D#.global_addr[31:0]
    g0[3] = ((unsigned)(ga >> 32) & 0x01FFFFFFu) | 0x80000000u;  // addr[56:32] | type=2
    i32x8 g1;
    // data_size=4B | pad_enable | pad_interval=32dw | pad_amount=8dw
    g1[0] = (int)((2u << 16) | (1u << 20) | (4u << 22) | (7u << 25));
    g1[1] = (int)(9600u << 16);                       // tensor_dim0 = 9600 elems
    g1[2] = (int)(1u << 16);                          // tensor_dim1 = 1
    g1[3] = (int)(9600u << 16);                       // tile_dim0 = 9600
    g1[4] = 0;                                        // tile_dim1/2 unused
    g1[5] = 9600;                                     // tensor_dim0_stride
    g1[6] = 0; g1[7] = 0;
    i32x4 z4 = {0, 0, 0, 0};
#ifdef TDM_6ARG
    i32x8 z8 = {0, 0, 0, 0, 0, 0, 0, 0};
    __builtin_amdgcn_tensor_load_to_lds(g0, g1, z4, z4, z8, 0);
#else
    __builtin_amdgcn_tensor_load_to_lds(g0, g1, z4, z4, 0);
#endif
    __builtin_amdgcn_s_wait_tensorcnt(0);
  }
  __syncthreads();
#else
  for (int i = threadIdx.x; i < NN*32; i += 608)
    sA[(i >> 5)*SASTR + (i & 31)] = vsrc[i];
  __syncthreads();
#endif

  const int wid = threadIdx.x >> 5, lane = threadIdx.x & 31;
  const int i0 = wid * 16;
  const int m = lane & 15, khalf = lane >> 4, col = lane & 15;
  const int i = i0 + m;
  const bool rowok = (i < NN);
  const int ic = rowok ? i : (NN - 1);
  const int phv = ph[bt*NN + ic];
  const float* __restrict__ srow = Sb + (size_t)phv*NN*NN + (size_t)ic*NN;
  __builtin_prefetch(srow, 0, 1);

  // ---- hop 1: A from masked support, B from sA ----
  v8f acc0 = {}, acc1 = {};
  for (int j0 = 0; j0 < NN; j0 += 4) {
    const int kk = j0 + 2*khalf;
    v2f a;
    a.x = rowok ? srow[kk]     : 0.0f;
    a.y = rowok ? srow[kk + 1] : 0.0f;
    v2f b0, b1;
    b0.x = sA[kk*SASTR + col];       b0.y = sA[(kk+1)*SASTR + col];
    b1.x = sA[kk*SASTR + 16 + col];  b1.y = sA[(kk+1)*SASTR + 16 + col];
    acc0 = wmma_f32(a, b0, acc0);
    acc1 = wmma_f32(a, b1, acc1);
  }
#pragma unroll
  for (int r = 0; r < 8; ++r) {
    const int row = i0 + r + 8*khalf;
    if (row < NN) {
      const float r0 = acc0[r] + sA[row*SASTR + col];        // +I residual
      const float r1 = acc1[r] + sA[row*SASTR + 16 + col];
      const size_t gi = (size_t)(bt*NN + row) * 32;
      wo1[gi + col]      = r0;
      wo1[gi + 16 + col] = r1;
      sB[row*SASTR + col]      = r0;
      sB[row*SASTR + 16 + col] = r1;
    }
  }
  __syncthreads();

  // ---- hop 2: same masked A rows, B/residual from sB ----
  v8f acc2 = {}, acc3 = {};
  for (int j0 = 0; j0 < NN; j0 += 4) {
    const int kk = j0 + 2*khalf;
    v2f a;
    a.x = rowok ? srow[kk]     : 0.0f;
    a.y = rowok ? srow[kk + 1] : 0.0f;
    v2f b0, b1;
    b0.x = sB[kk*SASTR + col];       b0.y = sB[(kk+1)*SASTR + col];
    b1.x = sB[kk*SASTR + 16 + col];  b1.y = sB[(kk+1)*SASTR + 16 + col];
    acc2 = wmma_f32(a, b0, acc2);
    acc3 = wmma_f32(a, b1, acc3);
  }
#pragma unroll
  for (int r = 0; r < 8; ++r) {
    const int row = i0 + r + 8*khalf;
    if (row < NN) {
      const size_t gi = (size_t)(bt*NN + row) * 32;
      wo2[gi + col]      = acc2[r] + sB[row*SASTR + col];
      wo2[gi + 16 + col] = acc3[r] + sB[row*SASTR + 16 + col];
    }
  }
}

// ---------------------------------------------------------------------------
// Kernel 4: h = Σ_k out_k @ Wl_k + bl ; blocks 0/5/6 are diag-scaled v0.
// Wl staged in LDS (stride-40). Writes pre-norm xo into d_out in (b,c,n,t).
// ---------------------------------------------------------------------------
__global__ void k_final(const float* __restrict__ v0, const float* __restrict__ w1,
                        const float* __restrict__ w2, const float* __restrict__ w3,
                        const float* __restrict__ w4, const float* __restrict__ Wl,
                        const float* __restrict__ bl, const float* __restrict__ x,
                        const int* __restrict__ ph, const int* __restrict__ adjph,
                        const float* __restrict__ adp, float* __restrict__ xo)
{
  __shared__ float sWl[224 * WLSTR];                  // [k][c], stride 40
  for (int i = threadIdx.x; i < 224*32; i += 256)
    sWl[(i >> 5)*WLSTR + (i & 31)] = Wl[i];
  __syncthreads();

  const int wid = blockIdx.x * 8 + (threadIdx.x >> 5);
  if (wid < NPOS/16) {
    const int lane = threadIdx.x & 31;
    const int m = lane & 15, khalf = lane >> 4, col = lane & 15;
    const int flat = wid * 16 + m;                    // this lane's A row
    const int n = flat % NN;
    const int phv = ph[flat];
    const float eq = (adjph[n*NN + n] == phv) ? 1.0f : 0.0f;
    const float d0 = 1.0f + eq;                       // A0 diagonal
    const float dA = adp[n*NN + n] * d0;              // Aad diagonal
    const float scl[7]  = { d0, 1.0f, 1.0f, 1.0f, 1.0f, dA, dA*dA };
    const float* srcs[7] = { v0, w1, w2, w3, w4, v0, v0 };
    v8f acc0 = {}, acc1 = {};
#pragma unroll
    for (int blk = 0; blk < 7; ++blk) {               // 56 K-steps, 112 WMMA/wave
      const float* arow = srcs[blk] + (size_t)flat * 32;
      const float s = scl[blk];
#pragma unroll
      for (int k0 = 0; k0 < 32; k0 += 4) {
        const int kk = k0 + 2*khalf;
        v2f a;  a.x = arow[kk] * s;  a.y = arow[kk + 1] * s;
        const int kg = blk*32 + kk;
        v2f b0, b1;
        b0.x = sWl[kg*WLSTR + col];       b0.y = sWl[(kg+1)*WLSTR + col];
        b1.x = sWl[kg*WLSTR + 16 + col];  b1.y = sWl[(kg+1)*WLSTR + 16 + col];
        acc0 = wmma_f32(a, b0, acc0);
        acc1 = wmma_f32(a, b1, acc1);
      }
    }
    const int c0 = col, c1 = col + 16;
    const float bl0 = bl[c0], bl1 = bl[c1];
    int rb = wid*16 + 8*khalf;
    int b2 = rb / (TCC*NN); int rr = rb % (TCC*NN); int t2 = rr / NN; int n2 = rr % NN;
#pragma unroll
    for (int r = 0; r < 8; ++r) {
      const size_t o0 = (((size_t)(b2*CR + c0))*NN + n2)*TCC + t2;
      const size_t o1 = (((size_t)(b2*CR + c1))*NN + n2)*TCC + t2;
      const size_t x0i = (((size_t)(b2*CR + c0))*NN + n2)*TTF + t2 + 1;  // x[..., -47:]
      const size_t x1i = (((size_t)(b2*CR + c1))*NN + n2)*TTF + t2 + 1;
      xo[o0] = acc0[r] + bl0 + x[x0i];
      xo[o1] = acc1[r] + bl1 + x[x1i];
      decode_step(b2, t2, n2);
    }
  }
}

// ---------------------------------------------------------------------------
// Kernel 5: BatchNorm batch stats (biased var) per channel
// ---------------------------------------------------------------------------
__global__ void k_bnstats(const float* __restrict__ xo, float* __restrict__ stats)
{
  const int c = blockIdx.x;
  __shared__ float ssum[256], ssq[256];
  float s = 0.0f, q = 0.0f;
  for (int bb2 = 0; bb2 < BB; ++bb2) {
    const float* base = xo + ((size_t)(bb2*CR + c)*NN)*TCC;
    for (int i = threadIdx.x; i < NN*TCC; i += 256) {
      const float v = base[i]; s += v; q += v*v;
    }
  }
  ssum[threadIdx.x] = s; ssq[threadIdx.x] = q;
  __syncthreads();
  for (int st = 128; st > 0; st >>= 1) {
    if (threadIdx.x < st) {
      ssum[threadIdx.x] += ssum[threadIdx.x + st];
      ssq[threadIdx.x]  += ssq[threadIdx.x + st];
    }
    __syncthreads();
  }
  if (threadIdx.x == 0) {
    const float cnt = (float)(BB*NN*TCC);
    const float mu  = ssum[0] / cnt;
    const float var = ssq[0] / cnt - mu*mu;
    stats[c*2]     = mu;
    stats[c*2 + 1] = rsqrtf(var + 1e-5f);
  }
}

// ---------------------------------------------------------------------------
// Kernel 6: normalize in place: (x-mu)*invstd*gamma + beta
// ---------------------------------------------------------------------------
__global__ void k_bnapply(float* __restrict__ xo, const float* __restrict__ stats,
                          const float* __restrict__ gamma, const float* __restrict__ beta)
{
  const size_t i = (size_t)blockIdx.x * 256 + threadIdx.x;   // XO_SZ exact
  const int c = (int)((i / (NN*TCC)) % CR);
  xo[i] = (xo[i] - stats[c*2]) * stats[c*2 + 1] * gamma[c] + beta[c];
}

// ---------------------------------------------------------------------------
extern "C" void kernel_launch(void* const* d_in, const int* in_sizes, int n_in,
                              void* d_out, int out_size, void* d_ws, size_t ws_size,
                              hipStream_t stream)
{
  (void)in_sizes; (void)n_in; (void)out_size; (void)ws_size;
  const float* x     = (const float*)d_in[0];
  const float* xph   = (const float*)d_in[1];
  const float* skip  = (const float*)d_in[2];
  const float* adp   = (const float*)d_in[3];
  const float* meanv = (const float*)d_in[4];
  const float* stdv  = (const float*)d_in[5];
  const float* s1    = (const float*)d_in[6];
  const float* s2    = (const float*)d_in[7];
  const float* Wf    = (const float*)d_in[8];
  const float* bf    = (const float*)d_in[9];
  const float* Wg    = (const float*)d_in[10];
  const float* bg    = (const float*)d_in[11];
  const float* Ws    = (const float*)d_in[12];
  const float* bs    = (const float*)d_in[13];
  const float* Wl    = (const float*)d_in[14];
  const float* bl    = (const float*)d_in[15];
  const float* gamma = (const float*)d_in[16];
  const float* beta  = (const float*)d_in[17];
  const int*   adjph = (const int*)d_in[18];

  float* xo    = (float*)d_out;          // output 0: (B,32,N,47)
  float* oskip = xo + XO_SZ;             // output 1: (B,256,N,47)

  float* ws = (float*)d_ws;
  float* v0 = ws;                        // (b,t,n,c)
  float* w1 = v0 + (size_t)NPOS*32;
  float* w2 = w1 + (size_t)NPOS*32;
  float* w3 = w2 + (size_t)NPOS*32;
  float* w4 = w3 + (size_t)NPOS*32;
  int*   ph = (int*)(w4 + (size_t)NPOS*32);
  float* stats = (float*)(ph + NPOS);
  float* Smask = stats + 64;             // 2 supports x 8 phases x 300x300

  k_tcn    <<<NPOS/8, 256, 0, stream>>>(x, xph, meanv, stdv, Wf, bf, Wg, bg, v0, ph);
  k_mask   <<<(2*NN*NN + 255)/256, 256, 0, stream>>>(s1, s2, adjph, Smask);
  k_skip   <<<NPOS/16, 256, 0, stream>>>(v0, Ws, bs, skip, oskip);
  {
    dim3 grid(BB*TCC, 2);
    k_hop2 <<<grid, 608, 0, stream>>>(Smask, ph, v0, w1, w2, w3, w4);
  }
  k_final  <<<(NPOS/16 + 7)/8, 256, 0, stream>>>(v0, w1, w2, w3, w4, Wl, bl, x, ph, adjph, adp, xo);
  k_bnstats<<<CR, 256, 0, stream>>>(xo, stats);
  k_bnapply<<<XO_SZ/256, 256, 0, stream>>>(xo, stats, gamma, beta);
}